// Grid_10093173146266
// MI455X (gfx1250) — compile-verified
//
#include <hip/hip_runtime.h>
#include <stdint.h>

typedef __attribute__((ext_vector_type(2))) float        v2f;
typedef __attribute__((ext_vector_type(4))) float        v4f;
typedef __attribute__((ext_vector_type(4))) unsigned int v4u;
typedef __attribute__((ext_vector_type(8))) int          v8i;
typedef __attribute__((ext_vector_type(4))) int          v4i;

#define N_POINTS  (1 << 20)
#define NUM_LVL   16
#define LDS_LVLS  5
#define MAX_ENTRY (1 << 19)
#define BLOCK     256

// res_i = floor(16 * 2^(7i/15)), exact values of the reference's double math
constexpr int kRes[NUM_LVL] = {16, 22, 30, 42, 58, 80, 111, 153,
                               212, 294, 406, 561, 776, 1072, 1482, 2048};
// prefix offsets (in float2 entries) of the LDS-resident levels 0..4
constexpr int kLdsOff[LDS_LVLS] = {0, 256, 740, 1640, 3404};
constexpr int kLdsTotal = 6768;   // 256+484+900+1764+3364 entries = 54144 B

struct Tabs { const v2f* t[NUM_LVL]; };

#if defined(__has_builtin)
#if __has_builtin(__builtin_amdgcn_tensor_load_to_lds) && \
    __has_builtin(__builtin_amdgcn_s_wait_tensorcnt)
#define HAVE_TDM 1
#endif
#endif

#if HAVE_TDM
// Issue one TDM descriptor copying `sz` float2 entries from `gsrc` to LDS
// byte-offset `laddr`.  D# packing per CDNA5 ISA:
// group0 = {count=1, lds_addr, global_addr[56:0], type=2}
// group1 = {data_size=3 (8B), tensor_dim0=sz, tensor_dim1=1, tile_dim0=sz,
//           tile_dim1=1, tensor_dim0_stride=sz}
__device__ __forceinline__ void tdm_copy_level(const v2f* gsrc, uint32_t laddr,
                                               int sz)
{
    const uint64_t gaddr = (uint64_t)(uintptr_t)gsrc;

    v4u g0 = {0u, 0u, 0u, 0u};
    g0.x = 1u;
    g0.y = laddr;
    g0.z = (uint32_t)(gaddr & 0xffffffffu);
    g0.w = ((uint32_t)((gaddr >> 32) & 0x01ffffffu)) | (2u << 30);

    v8i g1 = {0, 0, 0, 0, 0, 0, 0, 0};
    g1[0] = (int)(3u << 16);                                      // data_size = 8B
    g1[1] = (int)(((uint32_t)sz & 0xffffu) << 16);                // tensor_dim0 lo
    g1[2] = (int)((((uint32_t)sz >> 16) & 0xffffu) | (1u << 16)); // dim0 hi, dim1=1
    g1[3] = (int)(((uint32_t)sz & 0xffffu) << 16);                // tile_dim0 = sz
    g1[4] = 1;                                                    // tile_dim1 = 1
    g1[5] = sz;                                                   // dim0_stride

    v4i g2 = {0, 0, 0, 0};
    v4i g3 = {0, 0, 0, 0};
#if __clang_major__ >= 23
    v8i g4 = {0, 0, 0, 0, 0, 0, 0, 0};
    __builtin_amdgcn_tensor_load_to_lds(g0, g1, g2, g3, g4, 0);
#else
    __builtin_amdgcn_tensor_load_to_lds(g0, g1, g2, g3, 0);
#endif
}
#endif

__global__ __launch_bounds__(BLOCK)
void hashgrid_enc_kernel(const v2f* __restrict__ x, Tabs tabs,
                         v4f* __restrict__ out)
{
    __shared__ v2f sTab[kLdsTotal];

    // ---------------- stage coarse tables (levels 0..4) into LDS ------------
#if HAVE_TDM
    // TDM ops from the SAME wave are queued in order (ISA §7.1), so spread the
    // five level-DMAs across waves 0..4 for concurrent transfers.  Scalarize
    // the wave id so the gating is pure SALU.
    {
        const unsigned wv =
            (unsigned)__builtin_amdgcn_readfirstlane((int)(threadIdx.x >> 5));
#pragma unroll
        for (int l = 0; l < LDS_LVLS; ++l) {
            if (wv == (unsigned)l) {
                tdm_copy_level(tabs.t[l],
                               (uint32_t)(uintptr_t)&sTab[kLdsOff[l]],
                               kRes[l] * kRes[l]);
            }
        }
    }
    __builtin_amdgcn_s_wait_tensorcnt(0);  // each wave drains only its own DMA
    __syncthreads();
#else
#pragma unroll
    for (int l = 0; l < LDS_LVLS; ++l) {
        const v2f* src = tabs.t[l];
        const int sz = kRes[l] * kRes[l];
        for (int i = threadIdx.x; i < sz; i += BLOCK)
            sTab[kLdsOff[l] + i] = src[i];
    }
    __syncthreads();
#endif

    // ---------------- per-point encode --------------------------------------
    const int p = blockIdx.x * BLOCK + threadIdx.x;
    const v2f xy = __builtin_nontemporal_load(&x[p]);

    float outv[2 * NUM_LVL];

#pragma unroll
    for (int l = 0; l < NUM_LVL; ++l) {
        const int   res = kRes[l];
        const float rm1 = (float)(res - 1);
        const float hi  = (float)((double)res - 1.0001);   // reference clip bound

        float cx = fminf(fmaxf(xy.x * rm1, 0.0f), hi);
        float cy = fminf(fmaxf(xy.y * rm1, 0.0f), hi);
        float fx = floorf(cx), fy = floorf(cy);
        float dx = cx - fx,    dy = cy - fy;
        int   x0 = (int)fx,    y0 = (int)fy;

        v2f f00, f10, f01, f11;
        if (l < LDS_LVLS) {                        // dense grid resident in LDS
            const v2f* tb = &sTab[kLdsOff[l]];
            const int b = y0 * res + x0;
            f00 = tb[b];       f10 = tb[b + 1];
            f01 = tb[b + res]; f11 = tb[b + res + 1];
        } else if ((long)res * res < MAX_ENTRY) {  // dense grid in L2
            const v2f* tb = tabs.t[l];
            const int b = y0 * res + x0;
            f00 = tb[b];       f10 = tb[b + 1];
            f01 = tb[b + res]; f11 = tb[b + res + 1];
        } else {                                   // XOR-prime hashed level
            // (a*p ^ b*q) & (2^19-1): masking commutes with XOR, so 32-bit
            // products are bit-exact with the reference's int64 math.
            const v2f* tb = tabs.t[l];
            const uint32_t hx0 = (uint32_t)x0       * 3367900313u;
            const uint32_t hx1 = (uint32_t)(x0 + 1) * 3367900313u;
            const uint32_t hy0 = (uint32_t)y0       * 2654435761u;
            const uint32_t hy1 = (uint32_t)(y0 + 1) * 2654435761u;
            const uint32_t m = MAX_ENTRY - 1;
            f00 = tb[(hx0 ^ hy0) & m];
            f10 = tb[(hx1 ^ hy0) & m];
            f01 = tb[(hx0 ^ hy1) & m];
            f11 = tb[(hx1 ^ hy1) & m];
        }

        const float w00 = (1.0f - dx) * (1.0f - dy);
        const float w10 = dx * (1.0f - dy);
        const float w01 = (1.0f - dx) * dy;
        const float w11 = dx * dy;

        outv[2 * l]     = f00.x * w00 + f10.x * w10 + f01.x * w01 + f11.x * w11;
        outv[2 * l + 1] = f00.y * w00 + f10.y * w10 + f01.y * w01 + f11.y * w11;
    }

    // 128 B per point, streamed with non-temporal b128 stores (never re-read)
    v4f* orow = out + (size_t)p * 8;
#pragma unroll
    for (int k = 0; k < 8; ++k) {
        v4f v = {outv[4 * k], outv[4 * k + 1], outv[4 * k + 2], outv[4 * k + 3]};
        __builtin_nontemporal_store(v, &orow[k]);
    }
}

extern "C" void kernel_launch(void* const* d_in, const int* in_sizes, int n_in,
                              void* d_out, int out_size, void* d_ws, size_t ws_size,
                              hipStream_t stream)
{
    (void)in_sizes; (void)n_in; (void)out_size; (void)d_ws; (void)ws_size;
    const v2f* x = (const v2f*)d_in[0];
    Tabs tabs;
    for (int l = 0; l < NUM_LVL; ++l) tabs.t[l] = (const v2f*)d_in[1 + l];
    hashgrid_enc_kernel<<<N_POINTS / BLOCK, BLOCK, 0, stream>>>(
        x, tabs, (v4f*)d_out);
}